// SPE_34600256537189
// MI455X (gfx1250) — compile-verified
//
#include <hip/hip_runtime.h>
#include <hip/hip_bf16.h>

typedef float v2f __attribute__((ext_vector_type(2)));
typedef float v8f __attribute__((ext_vector_type(8)));

#define B_ 2
#define C_ 64
#define N_ 32768
#define K_ 16
#define O_ 64
#define BN_EPS 1e-5f

// Pass 1: transpose feature (B, C, N) -> pcT (B, N, C) so gathered columns are
// 256 contiguous bytes.
__global__ __launch_bounds__(256) void spe_transpose(const float* __restrict__ in,
                                                     float* __restrict__ out) {
  __shared__ float tile[64][65];
  const int b  = blockIdx.y;
  const int n0 = blockIdx.x * 64;
  for (int i = threadIdx.x; i < 64 * 64; i += 256) {
    const int c = i >> 6, j = i & 63;  // coalesced along n
    tile[c][j] = in[(size_t)b * C_ * N_ + (size_t)c * N_ + (n0 + j)];
  }
  __syncthreads();
  for (int i = threadIdx.x; i < 64 * 64; i += 256) {
    const int j = i >> 6, c = i & 63;  // coalesced along c
    out[(size_t)b * N_ * C_ + (size_t)(n0 + j) * C_ + c] = tile[c][j];
  }
}

// Pass 2: one wave per 16-point tile, cross-iteration double-buffered gathers.
//   D[o, col] = sum_c (inv[o]*W[o,c]) * X[c, col] + (beta[o] - mean[o]*inv[o])
//   S[o, col] += relu(D) over 17 columns (16 neighbors + center), then store.
__global__ __launch_bounds__(128) void spe_main(
    const float* __restrict__ pcT, const int* __restrict__ nidx,
    const float* __restrict__ W, const float* __restrict__ gamma,
    const float* __restrict__ beta, const float* __restrict__ rmean,
    const float* __restrict__ rvar, float* __restrict__ out) {
  __shared__ float Wf_lds[O_ * C_];   // 16 KB, fragment-major prescaled W'
  __shared__ float inv_lds[O_];
  __shared__ float shift_lds[O_];

  // ---- Per-output-channel BN terms (cooperative, once per block).
  for (int o = threadIdx.x; o < O_; o += 128) {
    const float iv = gamma[o] / sqrtf(rvar[o] + BN_EPS);
    inv_lds[o]   = iv;
    shift_lds[o] = beta[o] - rmean[o] * iv;
  }
  __syncthreads();

  // ---- Scale W and scatter into fragment-major layout:
  // pair slot ((t*4 + mt)*32 + L) holds W'[mt*16 + (L&15)][4t + 2*(L>>4) + e].
  for (int i = threadIdx.x; i < O_ * C_; i += 128) {
    const int r = i >> 6, c = i & 63;           // W row (o), col (c)
    const float w = W[i] * inv_lds[r];
    const int t  = c >> 2;
    const int hf = (c >> 1) & 1;
    const int e  = c & 1;
    const int L  = (hf << 4) | (r & 15);
    Wf_lds[((t * 4 + (r >> 4)) * 32 + L) * 2 + e] = w;
  }
  __syncthreads();

  const int lane = threadIdx.x & 31;
  const int wave = blockIdx.x * (blockDim.x >> 5) + (threadIdx.x >> 5);
  const int col  = lane & 15;   // point within tile
  const int half = lane >> 4;   // selects K pair {0,1} vs {2,3} in A/B frags
  const int b    = wave / (N_ / 16);
  const int n0   = (wave % (N_ / 16)) * 16;
  const int ctr  = n0 + col;    // center column index (17th column, k==16)

  // ---- BN shift as the WMMA C-operand (applied before ReLU, per column).
  v8f C0[4];
#pragma unroll
  for (int mt = 0; mt < 4; ++mt) {
#pragma unroll
    for (int v = 0; v < 8; ++v) C0[mt][v] = shift_lds[mt * 16 + half * 8 + v];
  }

  v8f S[4] = {};

  const int*   ip    = nidx + ((size_t)b * N_ + ctr) * K_;
  const float* base  = pcT + (size_t)b * N_ * C_ + half * 2;
  const float* afrag = &Wf_lds[lane * 2];  // A frag (t,mt) at +((t*4+mt)*64) floats

  // Issue the 16 gather loads for one column set (cidx) into Bf.
  // 32-bit unsigned offset (cidx*64 floats) keeps uniform-base GVS addressing.
  auto loadB = [&](v2f* Bf, int cidx) {
    const float* colp = base + ((uint32_t)cidx << 6);
#pragma unroll
    for (int t = 0; t < 16; ++t) Bf[t] = *(const v2f*)(colp + t * 4);
  };

  // 64-WMMA chain on a loaded column set, then exact ReLU-accumulate into S:
  // relu(x) = 0.5*(x + |x|)  (x+|x| is exactly 2x or 0; 0.5* is exact), which
  // avoids the canonicalize+max lowering of fmaxf.
  auto accum = [&](const v2f* Bf) {
    v8f D[4] = {C0[0], C0[1], C0[2], C0[3]};
#pragma unroll
    for (int t = 0; t < 16; ++t) {
#pragma unroll
      for (int mt = 0; mt < 4; ++mt) {
        const v2f Af = *(const v2f*)(afrag + (t * 4 + mt) * 64);
        D[mt] = __builtin_amdgcn_wmma_f32_16x16x4_f32(
            false, Af, false, Bf[t], (short)0, D[mt], false, false);
      }
    }
#pragma unroll
    for (int mt = 0; mt < 4; ++mt) {
#pragma unroll
      for (int v = 0; v < 8; ++v) {
        const float x = D[mt][v];
        S[mt][v] = __builtin_fmaf(0.5f, x + __builtin_fabsf(x), S[mt][v]);
      }
    }
  };

  // ---- Software pipeline: ping-pong BfA/BfB so the 16 gathers for column
  // set k+1 are in flight during the entire WMMA chain of column set k.
  v2f BfA[16], BfB[16];

  int i_next = ip[0];          // cidx for k=0
  loadB(BfA, i_next);
  i_next = ip[1];              // cidx for k=1

#pragma unroll 1
  for (int kk = 0; kk < 8; ++kk) {
    const int c_odd = i_next;                                    // cidx(2kk+1)
    const int c_ev2 = (2 * kk + 2 < 16) ? ip[2 * kk + 2] : ctr;  // cidx(2kk+2)
    loadB(BfB, c_odd);     // gathers for odd step in flight...
    accum(BfA);            // ...while computing the even step
    const int c_od2 = (2 * kk + 3 < 16) ? ip[2 * kk + 3] : ctr;  // cidx(2kk+3)
    loadB(BfA, c_ev2);     // gathers for next even step in flight...
    accum(BfB);            // ...while computing the odd step
    i_next = c_od2;
  }
  accum(BfA);  // tail: k=16, the center column (loaded in the last kk)

  // ---- Store out (B, O, N, 1): low half-lanes write 64B contiguous rows.
#pragma unroll
  for (int mt = 0; mt < 4; ++mt) {
#pragma unroll
    for (int v = 0; v < 8; ++v) {
      const int o = mt * 16 + half * 8 + v;
      out[((size_t)b * O_ + o) * N_ + ctr] = S[mt][v];
    }
  }
}

extern "C" void kernel_launch(void* const* d_in, const int* in_sizes, int n_in,
                              void* d_out, int out_size, void* d_ws, size_t ws_size,
                              hipStream_t stream) {
  const float* feature = (const float*)d_in[0];
  const int*   nidx    = (const int*)d_in[1];
  const float* W       = (const float*)d_in[2];
  const float* gamma   = (const float*)d_in[3];
  const float* beta    = (const float*)d_in[4];
  const float* rmean   = (const float*)d_in[5];
  const float* rvar    = (const float*)d_in[6];
  float* out = (float*)d_out;
  float* pcT = (float*)d_ws;  // needs B*N*C*4 = 16.8 MB

  dim3 g1(N_ / 64, B_);
  spe_transpose<<<g1, 256, 0, stream>>>(feature, pcT);

  const int nwaves = B_ * N_ / 16;  // 4096 tiles, 4 waves per 128-thread block
  spe_main<<<nwaves / 4, 128, 0, stream>>>(pcT, nidx, W, gamma, beta, rmean,
                                           rvar, out);
}